// Encoder_23905787969981
// MI455X (gfx1250) — compile-verified
//
#include <hip/hip_runtime.h>
#include <hip/hip_bf16.h>
#include <math.h>

// ---------------------------------------------------------------------------
// CDNA5 (gfx1250) transformer encoder, fp32 WMMA (V_WMMA_F32_16X16X4_F32).
// wave32; each wave owns 16x16 output tiles; C/D layout per ISA 7.12.2:
//   VGPR g : lanes 0-15 -> M=g, lanes 16-31 -> M=g+8 ; N = lane%16
// A (16x4) : lane<16: m=lane, k={kk,kk+1}; lane>=16: m=lane-16, k={kk+2,kk+3}
// B (4x16) : lane<16: n=lane, k={kk,kk+1}; lane>=16: n=lane-16, k={kk+2,kk+3}
// ---------------------------------------------------------------------------

typedef float v2f __attribute__((ext_vector_type(2)));
typedef float v8f __attribute__((ext_vector_type(8)));

#define ENC_B  128
#define ENC_D  256
#define ENC_H  4
#define ENC_DK 64
#define ENC_L  2
#define LN_EPS 1e-5f

__device__ __forceinline__ v8f wmma_f32_4(v2f a, v2f b, v8f c) {
  return __builtin_amdgcn_wmma_f32_16x16x4_f32(false, a, false, b, (short)0, c,
                                               false, false);
}

__device__ __forceinline__ float wave_sum(float s) {
  s += __shfl_xor(s, 1, 32);
  s += __shfl_xor(s, 2, 32);
  s += __shfl_xor(s, 4, 32);
  s += __shfl_xor(s, 8, 32);
  s += __shfl_xor(s, 16, 32);
  return s;
}

// K=256 WMMA inner product: acc[4] covers this wave's 16x64 slice.
// A fragments from LDS (stride 260, conflict-free); W direct from global
// (256 KB weights stay resident in the 192 MB L2).
__device__ __forceinline__ void gemm_inner(const float* __restrict__ Al,
                                           const float* __restrict__ W,
                                           int lane16, int half, int ntb,
                                           v8f (&acc)[4]) {
#pragma unroll 8
  for (int kk = 0; kk < 256; kk += 4) {
    int ka = kk + 2 * half;
    v2f a;
    a.x = Al[lane16 * 260 + ka];
    a.y = Al[lane16 * 260 + ka + 1];
    const float* w0 = W + (size_t)ka * 256;
#pragma unroll
    for (int nt = 0; nt < 4; ++nt) {
      int n = ntb + nt * 16 + lane16;
      v2f b;
      b.x = w0[n];
      b.y = w0[256 + n];
      acc[nt] = wmma_f32_4(a, b, acc[nt]);
    }
  }
}

// cooperative, coalesced stage of a 16x256 A tile into LDS (stride 260)
__device__ __forceinline__ void stage_A(float* __restrict__ Al,
                                        const float* __restrict__ A,
                                        int m0, int tid) {
  for (int e = tid; e < 4096; e += 128) {
    int row = e >> 8, col = e & 255;
    Al[row * 260 + col] = A[(size_t)(m0 + row) * 256 + col];
  }
}

// ---------------------------------------------------------------------------
// Fused Q/K/V projection: stage A tile once, run 3 weight matrices,
// scatter to [B,H,S,64]; Q pre-scaled by 1/sqrt(d_k).
// ---------------------------------------------------------------------------
__global__ void gemm_qkv(const float* __restrict__ A,
                         const float* __restrict__ Wq,
                         const float* __restrict__ Wk,
                         const float* __restrict__ Wv,
                         float* __restrict__ q, float* __restrict__ k,
                         float* __restrict__ v, int M, int S, float invT) {
  __shared__ float Al[16 * 260];
  const int lane = threadIdx.x, wv = threadIdx.y;
  const int tid = wv * 32 + lane;
  const int lane16 = lane & 15, half = lane >> 4;
  const int m0 = blockIdx.x * 16;
  const int ntb = wv * 64;

  stage_A(Al, A, m0, tid);
  __syncthreads();

#pragma unroll
  for (int mi = 0; mi < 3; ++mi) {
    const float* W = (mi == 0) ? Wq : (mi == 1) ? Wk : Wv;
    float* o       = (mi == 0) ? q  : (mi == 1) ? k  : v;
    float scale    = (mi == 0) ? invT : 1.0f;

    v8f acc[4];
#pragma unroll
    for (int nt = 0; nt < 4; ++nt) acc[nt] = v8f{};
    gemm_inner(Al, W, lane16, half, ntb, acc);

#pragma unroll
    for (int nt = 0; nt < 4; ++nt) {
      int n = ntb + nt * 16 + lane16;
      int h = n >> 6, d = n & 63;
#pragma unroll
      for (int g = 0; g < 8; ++g) {
        int r = m0 + g + 8 * half;
        int bb = r / S;
        int ss = r - bb * S;
        o[(((size_t)bb * ENC_H + h) * S + ss) * 64 + d] = acc[nt][g] * scale;
      }
    }
  }
}

// ---------------------------------------------------------------------------
// out = relu(A @ W + bias)   (FFN fc1)
// ---------------------------------------------------------------------------
__global__ void gemm_bias_relu(const float* __restrict__ A,
                               const float* __restrict__ W,
                               const float* __restrict__ bias,
                               float* __restrict__ out, int M) {
  __shared__ float Al[16 * 260];
  const int lane = threadIdx.x, wv = threadIdx.y;
  const int tid = wv * 32 + lane;
  const int lane16 = lane & 15, half = lane >> 4;
  const int m0 = blockIdx.x * 16;
  const int ntb = wv * 64;

  stage_A(Al, A, m0, tid);
  __syncthreads();

  v8f acc[4];
#pragma unroll
  for (int nt = 0; nt < 4; ++nt) acc[nt] = v8f{};
  gemm_inner(Al, W, lane16, half, ntb, acc);

#pragma unroll
  for (int nt = 0; nt < 4; ++nt) {
    int n = ntb + nt * 16 + lane16;
    float bi = bias[n];
#pragma unroll
    for (int g = 0; g < 8; ++g) {
      int r = m0 + g + 8 * half;
      out[(size_t)r * 256 + n] = fmaxf(acc[nt][g] + bi, 0.0f);
    }
  }
}

// ---------------------------------------------------------------------------
// out = LayerNorm(A @ W + bias + res) * gam + bet   (fully fused epilogue)
// Accumulators (+bias) bounce through the A-tile LDS (block owns whole rows),
// then each wave LayerNorms 4 complete rows with shfl reductions.
// ---------------------------------------------------------------------------
__global__ void gemm_ln(const float* __restrict__ A,
                        const float* __restrict__ W,
                        const float* __restrict__ bias,
                        const float* __restrict__ res,
                        const float* __restrict__ gam,
                        const float* __restrict__ bet,
                        float* __restrict__ out, int M) {
  __shared__ float Al[16 * 260];
  const int lane = threadIdx.x, wv = threadIdx.y;
  const int tid = wv * 32 + lane;
  const int lane16 = lane & 15, half = lane >> 4;
  const int m0 = blockIdx.x * 16;
  const int ntb = wv * 64;

  stage_A(Al, A, m0, tid);
  __syncthreads();

  v8f acc[4];
#pragma unroll
  for (int nt = 0; nt < 4; ++nt) acc[nt] = v8f{};
  gemm_inner(Al, W, lane16, half, ntb, acc);

  __syncthreads(); // all fragment reads of Al done before overwriting it
#pragma unroll
  for (int nt = 0; nt < 4; ++nt) {
    int n = ntb + nt * 16 + lane16;
    float bi = bias[n];
#pragma unroll
    for (int g = 0; g < 8; ++g) {
      Al[(g + 8 * half) * 260 + n] = acc[nt][g] + bi;
    }
  }
  __syncthreads();

  // each wave LayerNorms 4 full rows (256 cols, 8 elems/lane)
#pragma unroll
  for (int j = 0; j < 4; ++j) {
    int row = wv * 4 + j;
    int r = m0 + row;
    const float* pr = res + (size_t)r * 256;
    float xv[8];
    float s = 0.0f;
#pragma unroll
    for (int jj = 0; jj < 8; ++jj) {
      int i = lane + jj * 32;
      xv[jj] = Al[row * 260 + i] + pr[i];
      s += xv[jj];
    }
    float mean = wave_sum(s) * (1.0f / 256.0f);
    float vv = 0.0f;
#pragma unroll
    for (int jj = 0; jj < 8; ++jj) {
      float d = xv[jj] - mean;
      vv += d * d;
    }
    float rs = rsqrtf(wave_sum(vv) * (1.0f / 256.0f) + LN_EPS);
#pragma unroll
    for (int jj = 0; jj < 8; ++jj) {
      int i = lane + jj * 32;
      out[(size_t)r * 256 + i] = (xv[jj] - mean) * rs * gam[i] + bet[i];
    }
  }
}

// ---------------------------------------------------------------------------
// Flash-style gated attention.  q,k,v: [B,H,S,64] (q pre-scaled by 1/sqrt(dk)).
// out: [B,S,256].  One wave per 16-query tile; block = 4 waves.
// Q tiles live in LDS; K/V tiles staged cooperatively; fragment loads are
// chunked 4 k-steps deep so DS latency overlaps the WMMA accumulation chain.
// gate = tanh(attn_w[h][cat(r,c)]) applied AFTER the softmax normalizer.
// ---------------------------------------------------------------------------
__global__ void attn_kernel(const float* __restrict__ q,
                            const float* __restrict__ k,
                            const float* __restrict__ v,
                            const float* __restrict__ attn_w_l, // [H,6]
                            const int* __restrict__ Np, const int* __restrict__ Kp,
                            int S, int QT, int nblk,
                            float* __restrict__ out) {
  __shared__ float Ql[4 * 16 * 65];
  __shared__ float Kl[16 * 65];
  __shared__ float Vl[16 * 65];
  __shared__ float Pl[4 * 16 * 17];

  const int lane   = threadIdx.x;
  const int wv     = threadIdx.y;
  const int tid    = wv * 32 + lane;
  const int lane16 = lane & 15;
  const int half   = lane >> 4;

  const int bh  = blockIdx.x / nblk;
  const int blk = blockIdx.x - bh * nblk;
  const int b   = bh / ENC_H;
  const int h   = bh - b * ENC_H;
  const int qt  = blk * 4 + wv;
  const bool valid = (qt < QT);
  const int q0 = qt * 16;

  const float* qb = q + (size_t)bh * S * 64;
  const float* kb = k + (size_t)bh * S * 64;
  const float* vb = v + (size_t)bh * S * 64;

  // each wave stages its own 16x64 Q tile (published by first loop-top barrier)
  const int qbase = wv * 1040;
  for (int e = lane; e < 1024; e += 32) {
    int row = e >> 6, d = e & 63;
    int gq = q0 + row;
    Ql[qbase + row * 65 + d] = (valid && gq < S) ? qb[(size_t)gq * 64 + d] : 0.0f;
  }

  const int NN = *Np, KK = *Kp;
  const int NKv = NN * KK;
  float gt[6];
#pragma unroll
  for (int c = 0; c < 6; ++c) gt[c] = tanhf(attn_w_l[h * 6 + c]);

  float rowm[8], rowz[8];
#pragma unroll
  for (int g = 0; g < 8; ++g) { rowm[g] = -1e30f; rowz[g] = 0.0f; }
  v8f o[4];
#pragma unroll
  for (int nt = 0; nt < 4; ++nt) o[nt] = v8f{};

  const int KT = QT; // key tiles
  for (int kt = 0; kt < KT; ++kt) {
    __syncthreads(); // protect LDS reuse; also publishes Ql on iteration 0
    for (int e = tid; e < 1024; e += 128) {
      int key = e >> 6, d = e & 63;
      int gk = kt * 16 + key;
      float kvK = (gk < S) ? kb[(size_t)gk * 64 + d] : 0.0f;
      float kvV = (gk < S) ? vb[(size_t)gk * 64 + d] : 0.0f;
      Kl[key * 65 + d] = kvK;
      Vl[key * 65 + d] = kvV;
    }
    __syncthreads();

    if (valid) {
      // scores S[16q x 16k]: 16 WMMA k-steps over d_k, chunked 4-deep
      v8f sf = {};
#pragma unroll
      for (int ch = 0; ch < 4; ++ch) {
        v2f a[4], bb[4];
#pragma unroll
        for (int j = 0; j < 4; ++j) {
          int kd = (ch * 4 + j) * 4 + 2 * half;
          a[j].x  = Ql[qbase + lane16 * 65 + kd];
          a[j].y  = Ql[qbase + lane16 * 65 + kd + 1];
          bb[j].x = Kl[lane16 * 65 + kd];
          bb[j].y = Kl[lane16 * 65 + kd + 1];
        }
#pragma unroll
        for (int j = 0; j < 4; ++j) sf = wmma_f32_4(a[j], bb[j], sf);
      }

      const int c0 = kt * 16 + lane16; // global key index for this lane
      const bool cvalid = (c0 < S);
#pragma unroll
      for (int g = 0; g < 8; ++g) {
        float s = cvalid ? sf[g] : -1e30f;
        float mx = s;
        mx = fmaxf(mx, __shfl_xor(mx, 1, 32));
        mx = fmaxf(mx, __shfl_xor(mx, 2, 32));
        mx = fmaxf(mx, __shfl_xor(mx, 4, 32));
        mx = fmaxf(mx, __shfl_xor(mx, 8, 32));
        float nm = fmaxf(rowm[g], mx);
        float sc = __expf(rowm[g] - nm);
        float ee = __expf(s - nm); // masked -> 0
        float zs = ee;
        zs += __shfl_xor(zs, 1, 32);
        zs += __shfl_xor(zs, 2, 32);
        zs += __shfl_xor(zs, 4, 32);
        zs += __shfl_xor(zs, 8, 32);
        rowz[g] = rowz[g] * sc + zs;
        rowm[g] = nm;
#pragma unroll
        for (int nt = 0; nt < 4; ++nt) o[nt][g] *= sc;

        // block-structured gate category for (r, c0)
        int r = q0 + g + 8 * half;
        int cat;
        if (r < NKv) {
          if (c0 == NKv)                                cat = 3;
          else if (r == c0)                             cat = 0;
          else if (c0 < NKv && (r / KK) == (c0 / KK))   cat = 1;
          else                                          cat = 2;
        } else if (r == NKv) {
          cat = (c0 < NKv) ? 4 : 5;
        } else {
          cat = 2; // padded rows, never written
        }
        float p = gt[cat] * ee;
        Pl[wv * 272 + (g + 8 * half) * 17 + lane16] = p;
      }
    }
    __syncthreads(); // P visible for A-fragment layout reads

    if (valid) {
      // O += P(16x16) @ V(16x64); P fragments hoisted across the 4 N-tiles
      v2f pa[4];
#pragma unroll
      for (int j = 0; j < 4; ++j) {
        int kp = j * 4 + 2 * half;
        pa[j].x = Pl[wv * 272 + lane16 * 17 + kp];
        pa[j].y = Pl[wv * 272 + lane16 * 17 + kp + 1];
      }
#pragma unroll
      for (int nt = 0; nt < 4; ++nt) {
        v2f bb[4];
#pragma unroll
        for (int j = 0; j < 4; ++j) {
          int kp = j * 4 + 2 * half;
          bb[j].x = Vl[kp * 65 + nt * 16 + lane16];
          bb[j].y = Vl[(kp + 1) * 65 + nt * 16 + lane16];
        }
        v8f oc = o[nt];
#pragma unroll
        for (int j = 0; j < 4; ++j) oc = wmma_f32_4(pa[j], bb[j], oc);
        o[nt] = oc;
      }
    }
  }

  if (valid) {
#pragma unroll
    for (int nt = 0; nt < 4; ++nt) {
#pragma unroll
      for (int g = 0; g < 8; ++g) {
        int r = q0 + g + 8 * half;
        if (r < S) {
          out[((size_t)b * S + r) * 256 + h * 64 + nt * 16 + lane16] =
              o[nt][g] / rowz[g];
        }
      }
    }
  }
}

// ---------------------------------------------------------------------------
// Host orchestration. Workspace: 5 x [B,S,D] fp32 buffers (~330 MB).
// Per layer: fused QKV -> attention -> fused mha_fc+LN -> fc1(relu)
//            -> fused fc2+LN ; then fused out_fc+LN into d_out.
// ---------------------------------------------------------------------------
extern "C" void kernel_launch(void* const* d_in, const int* in_sizes, int n_in,
                              void* d_out, int out_size, void* d_ws, size_t ws_size,
                              hipStream_t stream) {
  const float* samples  = (const float*)d_in[0];
  const float* Wq       = (const float*)d_in[1];
  const float* Wk       = (const float*)d_in[2];
  const float* Wv       = (const float*)d_in[3];
  const float* attn_w   = (const float*)d_in[4];
  const float* mha_fc_w = (const float*)d_in[5];
  const float* mha_fc_b = (const float*)d_in[6];
  const float* mha_ln_g = (const float*)d_in[7];
  const float* mha_ln_b = (const float*)d_in[8];
  const float* d_fc1_w  = (const float*)d_in[9];
  const float* d_fc1_b  = (const float*)d_in[10];
  const float* d_fc2_w  = (const float*)d_in[11];
  const float* d_fc2_b  = (const float*)d_in[12];
  const float* d_ln_g   = (const float*)d_in[13];
  const float* d_ln_b   = (const float*)d_in[14];
  const float* out_fc_w = (const float*)d_in[15];
  const float* out_fc_b = (const float*)d_in[16];
  const float* out_ln_g = (const float*)d_in[17];
  const float* out_ln_b = (const float*)d_in[18];
  const int*   Np       = (const int*)d_in[19];
  const int*   Kp       = (const int*)d_in[20];

  const int S = in_sizes[0] / (ENC_B * ENC_D); // 501
  const int M = ENC_B * S;                     // 64128 (multiple of 16)
  const size_t buf = (size_t)M * ENC_D;

  float* ws = (float*)d_ws;
  float* x  = ws;
  float* qb = ws + buf;
  float* kb = ws + 2 * buf;
  float* vb = ws + 3 * buf;
  float* t  = ws + 4 * buf;

  const int gemmGrid = M / 16;  // one block per 16-row tile, 4 waves cover N
  const dim3 gB(32, 4);
  const int QT   = (S + 15) / 16;
  const int nblk = (QT + 3) / 4;
  const int attnGrid = ENC_B * ENC_H * nblk;
  const float invT = 0.125f; // 1/sqrt(d_k)

  const float* xin = samples;
  for (int l = 0; l < ENC_L; ++l) {
    const size_t wOff = (size_t)l * ENC_D * 256;
    // fused Q/K/V projection -> [B,H,S,64] each (Q scaled by 1/sqrt(dk))
    gemm_qkv<<<gemmGrid, gB, 0, stream>>>(xin, Wq + wOff, Wk + wOff, Wv + wOff,
                                          qb, kb, vb, M, S, invT);
    // gated flash attention -> t [B,S,256]
    attn_kernel<<<attnGrid, gB, 0, stream>>>(qb, kb, vb, attn_w + (size_t)l * ENC_H * 6,
                                             Np, Kp, S, QT, nblk, t);
    // x = LN(t @ mha_fc_w + bias + residual)
    gemm_ln<<<gemmGrid, gB, 0, stream>>>(t, mha_fc_w + wOff, mha_fc_b + l * ENC_D,
                                         xin, mha_ln_g + l * ENC_D, mha_ln_b + l * ENC_D,
                                         x, M);
    // t = relu(x @ fc1 + b1)
    gemm_bias_relu<<<gemmGrid, gB, 0, stream>>>(x, d_fc1_w + wOff, d_fc1_b + l * ENC_D, t, M);
    // x = LN(t @ fc2 + b2 + x)
    gemm_ln<<<gemmGrid, gB, 0, stream>>>(t, d_fc2_w + wOff, d_fc2_b + l * ENC_D,
                                         x, d_ln_g + l * ENC_D, d_ln_b + l * ENC_D,
                                         x, M);
    xin = x;
  }
  // d_out = LN(x @ out_fc + bias + original samples)
  gemm_ln<<<gemmGrid, gB, 0, stream>>>(x, out_fc_w, out_fc_b,
                                       samples, out_ln_g, out_ln_b,
                                       (float*)d_out, M);
}